// SuperDCFrontShareLayer_19335942767070
// MI455X (gfx1250) — compile-verified
//
#include <hip/hip_runtime.h>
#include <math.h>

#ifndef __has_builtin
#define __has_builtin(x) 0
#endif

#define ROW_LEN 1024           // floats per input row
#define PAIRS   512            // complex pairs per row
#define TILE    256            // threads per block; block handles 256 pairs (half a row)

typedef int v2i __attribute__((ext_vector_type(2)));
typedef __attribute__((address_space(1))) int global_i32;
typedef __attribute__((address_space(3))) int lds_i32;
typedef __attribute__((address_space(1))) v2i global_v2i;
typedef __attribute__((address_space(3))) v2i lds_v2i;

#if __has_builtin(__builtin_amdgcn_global_load_async_to_lds_b32) && \
    __has_builtin(__builtin_amdgcn_global_load_async_to_lds_b64) && \
    __has_builtin(__builtin_amdgcn_s_wait_asynccnt)
#define USE_ASYNC 1
#else
#define USE_ASYNC 0
#endif

// Output complex index c per row:
//   c==0 -> (x[0],0);  c==1023 -> (x[1023],0)
//   else m=c-1, j=m>>1, t=w[j], k=sqrt(1-t^2+1e-6):
//        re = t*x[c];  im = k*x[c+1] (m even) | k*x[c-1] (m odd)
// Thread handles even pair (c0=2i, c1=2i+1)  ->  one aligned float4 store.
__global__ __launch_bounds__(TILE) void superdc_front_share_kernel(
    const float* __restrict__ x,
    const float* __restrict__ w,
    float4* __restrict__ out4)
{
    __shared__ __align__(16) float xs[516];  // xs[2..513] = x[2*i0 .. 2*i0+511]; halos xs[1], xs[514]
    __shared__ float ws[258];                // ws[1..256] = w[i0 .. i0+255];     halo ws[0]

    const int      tid = threadIdx.x;
    const unsigned blk = blockIdx.x;
    const int      row = (int)(blk >> 1);            // 2 half-rows per row
    const int      i0  = (int)(blk & 1u) * TILE;     // first pair index of this block

    const float* xrow = x + (size_t)row * ROW_LEN;

#if USE_ASYNC
    // CDNA5 async global->LDS staging (ASYNCcnt): 8B of x + 4B of w per lane.
    __builtin_amdgcn_global_load_async_to_lds_b64(
        (global_v2i*)(xrow + 2 * i0 + 2 * tid),
        (lds_v2i*)(xs + 2 + 2 * tid), 0, 0);
    if (i0 + tid < 511) {
        __builtin_amdgcn_global_load_async_to_lds_b32(
            (global_i32*)(w + i0 + tid),
            (lds_i32*)(ws + 1 + tid), 0, 0);
    }
    if (tid == 0 && i0 != 0) {                       // left halos
        __builtin_amdgcn_global_load_async_to_lds_b32(
            (global_i32*)(xrow + 2 * i0 - 1), (lds_i32*)(xs + 1), 0, 0);
        __builtin_amdgcn_global_load_async_to_lds_b32(
            (global_i32*)(w + i0 - 1), (lds_i32*)(ws + 0), 0, 0);
    }
    if (tid == TILE - 1 && i0 == 0) {                // right x halo: x[512]
        __builtin_amdgcn_global_load_async_to_lds_b32(
            (global_i32*)(xrow + 2 * TILE), (lds_i32*)(xs + 2 + 2 * TILE), 0, 0);
    }
    __builtin_amdgcn_s_wait_asynccnt(0);             // this wave's async copies done
#else
    xs[2 + 2 * tid]     = xrow[2 * i0 + 2 * tid];
    xs[2 + 2 * tid + 1] = xrow[2 * i0 + 2 * tid + 1];
    if (i0 + tid < 511)        ws[1 + tid] = w[i0 + tid];
    if (tid == 0 && i0 != 0) { xs[1] = xrow[2 * i0 - 1]; ws[0] = w[i0 - 1]; }
    if (tid == TILE - 1 && i0 == 0) xs[2 + 2 * TILE] = xrow[2 * TILE];
#endif
    __syncthreads();   // every wave passed its own wait -> whole tile valid

    const int   i  = i0 + tid;
    const int   lx = 2 + 2 * tid;
    const float x0 = xs[lx];       // x[2i]
    const float x1 = xs[lx + 1];   // x[2i+1]

    float4 o;
    if (i == 0) {                          // c=0 endpoint
        o.x = x0; o.y = 0.0f;
    } else {                               // c=2i, m odd, j=i-1
        const float t0 = ws[tid];
        const float k0 = __builtin_sqrtf(1.0f - t0 * t0 + 1e-6f);
        o.x = t0 * x0;
        o.y = k0 * xs[lx - 1];             // x[2i-1]
    }
    if (i == PAIRS - 1) {                  // c=1023 endpoint
        o.z = x1; o.w = 0.0f;
    } else {                               // c=2i+1, m even, j=i
        const float t1 = ws[tid + 1];
        const float k1 = __builtin_sqrtf(1.0f - t1 * t1 + 1e-6f);
        o.z = t1 * x1;
        o.w = k1 * xs[lx + 2];             // x[2i+2]
    }
    out4[(size_t)row * PAIRS + i] = o;     // 16B aligned, global_store_b128
}

extern "C" void kernel_launch(void* const* d_in, const int* in_sizes, int n_in,
                              void* d_out, int out_size, void* d_ws, size_t ws_size,
                              hipStream_t stream) {
    const float* x = (const float*)d_in[0];   // 32768 x 1024 f32
    const float* w = (const float*)d_in[1];   // 511 f32
    // d_in[2] = sample_arch (always 511; layout constants are compile-time)
    float4* out4 = (float4*)d_out;            // 32768 x 1024 complex64 -> 512 float4 per row

    const int rows = in_sizes[0] / ROW_LEN;   // 32768
    dim3 grid((unsigned)(rows * 2));          // 2 half-row blocks per row
    dim3 block(TILE);
    superdc_front_share_kernel<<<grid, block, 0, stream>>>(x, w, out4);
}